// NodeNet_14276471292018
// MI455X (gfx1250) — compile-verified
//
#include <hip/hip_runtime.h>
#include <math.h>

// ---------------------------------------------------------------------------
// MI455X plan:
//  K1: bo = Ro^T X, bi = Ri^T X via V_WMMA_F32_16X16X4_F32, branch-free,
//      two independent WMMA chains per output. HBM-bound: 256 MB.
//  K2: mi = (Ri*e)@bo, mo = (Ro*e)@bi, per-node coalesced reduction. 256 MB.
//  K3: 12-wire circuit in VGPRs, one wave per state, i = (r<<5)|lane.
//      Optimized wire->bit permutation: lane bits carry wires {0,3,4,7,11}
//      (the least-rotated set), register bits carry {1,2,5,6,8,9,10}.
//      => 8 lane-RYs (swizzle) instead of 16, 9 of 16 CNOTs become free
//      register renames. Swizzles batched x8 to amortize s_wait_dscnt.
// ---------------------------------------------------------------------------

#define N_NODES 4096
#define N_EDGES 8192

typedef __attribute__((ext_vector_type(2))) float v2f;
typedef __attribute__((ext_vector_type(8))) float v8f;

// ---------------------------------------------------------------------------
// Kernel 1: bo = Ro^T @ X, bi = Ri^T @ X   (8192x4 each)
// ---------------------------------------------------------------------------
__global__ __launch_bounds__(128) void k_bt_x(const float* __restrict__ X,
                                              const float* __restrict__ Ri,
                                              const float* __restrict__ Ro,
                                              float* __restrict__ bi,
                                              float* __restrict__ bo) {
  const int lane = threadIdx.x & 31;
  const int tile = blockIdx.x * 4 + (threadIdx.x >> 5);   // 512 tiles of 16 edges
  const int e0   = tile * 16;
  const int m    = lane & 15;   // A row (edge) / B-D column
  const int hi   = lane >> 4;   // lane half -> K pair {0,1} vs {2,3}
  const int mb   = m & 3;       // wrapped feature idx: cols>=4 garbage, discarded

  v8f accO0 = {}, accO1 = {}, accI0 = {}, accI1 = {};

  for (int n0 = 0; n0 < N_NODES; n0 += 8) {
    {
      const int r0 = n0 + (hi ? 2 : 0);
      const int r1 = n0 + (hi ? 3 : 1);
      v2f a_o, a_i, b;
      a_o.x = Ro[(size_t)r0 * N_EDGES + e0 + m];
      a_o.y = Ro[(size_t)r1 * N_EDGES + e0 + m];
      a_i.x = Ri[(size_t)r0 * N_EDGES + e0 + m];
      a_i.y = Ri[(size_t)r1 * N_EDGES + e0 + m];
      b.x = X[r0 * 4 + mb];
      b.y = X[r1 * 4 + mb];
      accO0 = __builtin_amdgcn_wmma_f32_16x16x4_f32(false, a_o, false, b,
                                                    (short)0, accO0, false, false);
      accI0 = __builtin_amdgcn_wmma_f32_16x16x4_f32(false, a_i, false, b,
                                                    (short)0, accI0, false, false);
    }
    {
      const int r0 = n0 + 4 + (hi ? 2 : 0);
      const int r1 = n0 + 4 + (hi ? 3 : 1);
      v2f a_o, a_i, b;
      a_o.x = Ro[(size_t)r0 * N_EDGES + e0 + m];
      a_o.y = Ro[(size_t)r1 * N_EDGES + e0 + m];
      a_i.x = Ri[(size_t)r0 * N_EDGES + e0 + m];
      a_i.y = Ri[(size_t)r1 * N_EDGES + e0 + m];
      b.x = X[r0 * 4 + mb];
      b.y = X[r1 * 4 + mb];
      accO1 = __builtin_amdgcn_wmma_f32_16x16x4_f32(false, a_o, false, b,
                                                    (short)0, accO1, false, false);
      accI1 = __builtin_amdgcn_wmma_f32_16x16x4_f32(false, a_i, false, b,
                                                    (short)0, accI1, false, false);
    }
  }

  const v8f accO = accO0 + accO1;
  const v8f accI = accI0 + accI1;

  if (m < 4) {
#pragma unroll
    for (int r = 0; r < 8; ++r) {
      const int row = e0 + r + (hi ? 8 : 0);
      bo[row * 4 + m] = accO[r];
      bi[row * 4 + m] = accI[r];
    }
  }
}

// ---------------------------------------------------------------------------
// Kernel 2: per-node coalesced reduction for mi/mo.
// ---------------------------------------------------------------------------
__global__ __launch_bounds__(256) void k_msg(const float* __restrict__ e,
                                             const float* __restrict__ Ri,
                                             const float* __restrict__ Ro,
                                             const float4* __restrict__ bo,
                                             const float4* __restrict__ bi,
                                             float* __restrict__ mi,
                                             float* __restrict__ mo) {
  __shared__ float red[256 * 8];
  const int n   = blockIdx.x;
  const int tid = threadIdx.x;

  float ai0 = 0.f, ai1 = 0.f, ai2 = 0.f, ai3 = 0.f;
  float ao0 = 0.f, ao1 = 0.f, ao2 = 0.f, ao3 = 0.f;

  for (int k = tid; k < N_EDGES; k += 256) {
    const float ek = e[k];
    const float ri = Ri[(size_t)n * N_EDGES + k] * ek;
    const float ro = Ro[(size_t)n * N_EDGES + k] * ek;
    const float4 bok = bo[k];
    const float4 bik = bi[k];
    ai0 += ri * bok.x; ai1 += ri * bok.y; ai2 += ri * bok.z; ai3 += ri * bok.w;
    ao0 += ro * bik.x; ao1 += ro * bik.y; ao2 += ro * bik.z; ao3 += ro * bik.w;
  }

  red[tid * 8 + 0] = ai0; red[tid * 8 + 1] = ai1;
  red[tid * 8 + 2] = ai2; red[tid * 8 + 3] = ai3;
  red[tid * 8 + 4] = ao0; red[tid * 8 + 5] = ao1;
  red[tid * 8 + 6] = ao2; red[tid * 8 + 7] = ao3;
  __syncthreads();

  for (int off = 128; off > 0; off >>= 1) {
    if (tid < off) {
#pragma unroll
      for (int f = 0; f < 8; ++f)
        red[tid * 8 + f] += red[(tid + off) * 8 + f];
    }
    __syncthreads();
  }

  if (tid < 4) mi[n * 4 + tid] = red[tid];
  if (tid >= 4 && tid < 8) mo[n * 4 + (tid - 4)] = red[tid];
}

// ---------------------------------------------------------------------------
// Kernel 3: circuit in VGPRs with optimized wire->bit permutation.
//   WBIT[w] = bit carrying wire w. Bits 0..4 = lane bits, 5..11 = reg bits.
//   Lane wires {0,3,4,7,11}; register wires (bit order 5..11) {1,2,5,6,8,9,10}.
// ---------------------------------------------------------------------------
constexpr int WBIT[12]  = {0, 5, 6, 1, 2, 7, 8, 3, 9, 10, 11, 4};
constexpr int LANEW[5]  = {0, 3, 4, 7, 11};
constexpr int REGW[7]   = {1, 2, 5, 6, 8, 9, 10};

template <int b>
__device__ __forceinline__ void ry_gate(float* v, int lane, float c, float s) {
  if constexpr (b >= 5) {
    constexpr int rb = b - 5;
#pragma unroll
    for (int r = 0; r < 128; ++r) {
      if ((r >> rb) & 1) continue;            // folds at compile time
      const int r1 = r | (1 << rb);
      const float a0 = v[r], a1 = v[r1];
      v[r]  = c * a0 - s * a1;
      v[r1] = s * a0 + c * a1;
    }
  } else {
    // new = c*v + sgn*s*partner ; sgn = +1 on the high half, -1 on the low.
    const float sgn = ((lane >> b) & 1) ? s : -s;
#pragma unroll
    for (int r0 = 0; r0 < 128; r0 += 8) {
      float ap[8];
#pragma unroll
      for (int j = 0; j < 8; ++j) ap[j] = __shfl_xor(v[r0 + j], 1 << b, 32);
#pragma unroll
      for (int j = 0; j < 8; ++j) v[r0 + j] = fmaf(c, v[r0 + j], sgn * ap[j]);
    }
  }
}

template <int bc, int bt>
__device__ __forceinline__ void cx_gate(float* v, int lane) {
  if constexpr (bc >= 5 && bt >= 5) {
    // pure register permutation; compiler renames it away
    constexpr int rc = bc - 5, rt = bt - 5;
#pragma unroll
    for (int r = 0; r < 128; ++r) {
      if (!((r >> rc) & 1) || ((r >> rt) & 1)) continue;
      const int r2 = r ^ (1 << rt);
      const float t = v[r]; v[r] = v[r2]; v[r2] = t;
    }
  } else if constexpr (bc >= 5 && bt < 5) {
    constexpr int rc = bc - 5;
#pragma unroll
    for (int r0 = 0; r0 < 128; r0 += 8) {
      float ap[8];
#pragma unroll
      for (int j = 0; j < 8; ++j) {
        if (((r0 + j) >> rc) & 1) ap[j] = __shfl_xor(v[r0 + j], 1 << bt, 32);
      }
#pragma unroll
      for (int j = 0; j < 8; ++j) {
        if (((r0 + j) >> rc) & 1) v[r0 + j] = ap[j];
      }
    }
  } else if constexpr (bc < 5 && bt >= 5) {
    constexpr int rt = bt - 5;
    const bool p = (lane >> bc) & 1;
#pragma unroll
    for (int r = 0; r < 128; ++r) {
      if ((r >> rt) & 1) continue;
      const int r2 = r | (1 << rt);
      const float a = v[r], b2 = v[r2];
      v[r]  = p ? b2 : a;
      v[r2] = p ? a : b2;
    }
  } else {
    const bool p = (lane >> bc) & 1;
#pragma unroll
    for (int r0 = 0; r0 < 128; r0 += 8) {
      float ap[8];
#pragma unroll
      for (int j = 0; j < 8; ++j) ap[j] = __shfl_xor(v[r0 + j], 1 << bt, 32);
#pragma unroll
      for (int j = 0; j < 8; ++j) v[r0 + j] = p ? ap[j] : v[r0 + j];
    }
  }
}

#define RY(a, w)                                                       \
  do {                                                                 \
    const float c_ = ((a) < 32) ? __shfl(thc, (a), 32) : t32c;         \
    const float s_ = ((a) < 32) ? __shfl(ths, (a), 32) : t32s;         \
    ry_gate<WBIT[(w)]>(v, lane, c_, s_);                               \
  } while (0)
#define CX(cw, tw) cx_gate<WBIT[(cw)], WBIT[(tw)]>(v, lane)

__global__ __launch_bounds__(256) void k_circuit(const float* __restrict__ mi,
                                                 const float* __restrict__ mo,
                                                 const float* __restrict__ X,
                                                 const float* __restrict__ theta,
                                                 float* __restrict__ out) {
  const int lane = threadIdx.x & 31;
  const int node = blockIdx.x * 8 + (threadIdx.x >> 5);   // one wave per state

  // --- distributed rotation coefficients ---------------------------------
  const float thAng = theta[lane] * 0.5f;                 // lane l: theta[l]
  const float thc = cosf(thAng), ths = sinf(thAng);
  const float a32 = theta[32] * 0.5f;                     // uniform
  const float t32c = cosf(a32), t32s = sinf(a32);
  float wAng = 0.0f;                                      // lanes 0..11: data angles
  if (lane < 12) {
    float ang;
    if (lane < 4)      ang = mi[node * 4 + lane];
    else if (lane < 8) ang = mo[node * 4 + (lane - 4)];
    else               ang = X[node * 4 + (lane - 8)];
    wAng = ang * 0.5f;
  }
  const float wic = cosf(wAng), wis = sinf(wAng);

  // --- analytic product-state init ---------------------------------------
  // amplitude(i) = prod_w (bit_{WBIT[w]}(i) ? sin_w : cos_w)
  float lp = 1.0f;                      // lane-bit wires
#pragma unroll
  for (int t = 0; t < 5; ++t) {
    const int w = LANEW[t];
    const float cw = __shfl(wic, w, 32);
    const float sw = __shfl(wis, w, 32);
    lp *= ((lane >> WBIT[w]) & 1) ? sw : cw;
  }
  float v[128];
  v[0] = lp;                            // register-bit wires via doubling
#pragma unroll
  for (int bit = 0; bit < 7; ++bit) {
    const int w = REGW[bit];
    const float cw = __shfl(wic, w, 32);
    const float sw = __shfl(wis, w, 32);
#pragma unroll
    for (int r = 0; r < (1 << bit); ++r) {
      v[r + (1 << bit)] = v[r] * sw;
      v[r]              = v[r] * cw;
    }
  }

  // --- the 49 fixed gates -------------------------------------------------
  RY(0, 1);  RY(1, 2);  CX(1, 2);
  RY(2, 3);  RY(3, 4);  CX(4, 3);
  RY(4, 5);  RY(5, 6);  CX(5, 6);
  RY(6, 7);  RY(7, 8);  CX(7, 8);
  RY(8, 9);  RY(9, 10); CX(10, 9);
  RY(10, 0); RY(11, 1); CX(0, 1);
  RY(12, 2); RY(13, 3); CX(3, 2);
  RY(14, 4); RY(15, 5); CX(4, 5);
  RY(16, 6); RY(17, 7); CX(7, 6);
  RY(18, 8); RY(19, 9); CX(8, 9);
  RY(20, 10); RY(21, 11); CX(11, 10);
  RY(22, 2); RY(23, 5); CX(5, 2);
  RY(24, 6); RY(25, 9); CX(6, 9);
  RY(26, 1); RY(27, 2); CX(1, 2);
  RY(28, 9); RY(29, 10); CX(10, 9);
  RY(30, 2); RY(31, 9); CX(2, 9);
  RY(32, 9);

  // --- <Z_9>: wire 9 -> bit 10 -> register bit 5; sign folds per-register --
  float acc = 0.0f;
#pragma unroll
  for (int r = 0; r < 128; ++r) {
    const float sq = v[r] * v[r];
    acc = ((r >> 5) & 1) ? (acc - sq) : (acc + sq);
  }
#pragma unroll
  for (int off = 16; off > 0; off >>= 1)
    acc += __shfl_xor(acc, off, 32);
  if (lane == 0)
    out[node] = 3.14159265358979323846f * (1.0f - acc) * 0.5f;
}

// ---------------------------------------------------------------------------
extern "C" void kernel_launch(void* const* d_in, const int* in_sizes, int n_in,
                              void* d_out, int out_size, void* d_ws, size_t ws_size,
                              hipStream_t stream) {
  const float* X  = (const float*)d_in[0];
  const float* e  = (const float*)d_in[1];
  const float* Ri = (const float*)d_in[2];
  const float* Ro = (const float*)d_in[3];
  const float* th = (const float*)d_in[4];
  float* out = (float*)d_out;

  char* ws = (char*)d_ws;
  float* bo = (float*)(ws);
  float* bi = (float*)(ws + (size_t)N_EDGES * 4 * sizeof(float));
  float* mi = (float*)(ws + (size_t)2 * N_EDGES * 4 * sizeof(float));
  float* mo = (float*)(ws + (size_t)2 * N_EDGES * 4 * sizeof(float)
                          + (size_t)N_NODES * 4 * sizeof(float));

  // 512 edge tiles, 4 waves (tiles) per 128-thread block.
  k_bt_x<<<128, 128, 0, stream>>>(X, Ri, Ro, bi, bo);
  // One block per node.
  k_msg<<<N_NODES, 256, 0, stream>>>(e, Ri, Ro,
                                     (const float4*)bo, (const float4*)bi,
                                     mi, mo);
  // One wave per state, 8 states per block.
  k_circuit<<<N_NODES / 8, 256, 0, stream>>>(mi, mo, X, th, out);
}